// BilinearInteractionLayer_64793876628087
// MI455X (gfx1250) — compile-verified
//
#include <hip/hip_runtime.h>

// BilinearInteractionLayer: out[b,p,:] = (x[b,i_p,:] @ W) * x[b,j_p,:], p over 496 (i<j) pairs.
// B=8192, F=32, E=64.  Write-bandwidth bound (~1.04 GB out, floor ~47us @ 23.3 TB/s).
// One block (8 wave32s) per batch element:
//   1) stage x[b] (8KB, non-temporal: read-once) + W (16KB, regular: L2-hot) into LDS
//   2) xw = x[b] @ W via V_WMMA_F32_16X16X4_F32: 8 waves x one 16x16 tile, K-loop of 16
//   3) stream 496 pair products with coalesced non-temporal b128 stores (16 pairs/iter, 31 iters)

#define NF 32
#define NE 64
#define NPAIRS 496   // 32*31/2

typedef float v2f __attribute__((ext_vector_type(2)));
typedef float v4f __attribute__((ext_vector_type(4)));
typedef float v8f __attribute__((ext_vector_type(8)));

__global__ __launch_bounds__(256)
void bilinear_wmma_kernel(const float* __restrict__ x,
                          const float* __restrict__ W,
                          float* __restrict__ out)
{
    __shared__ float lds_x [NF * NE];   // 8 KB: raw fields of this batch element
    __shared__ float lds_xw[NF * NE];   // 8 KB: x[b] @ W
    __shared__ float lds_W [NE * NE];   // 16 KB: weight matrix
    __shared__ unsigned short pair_i[NPAIRS];
    __shared__ unsigned short pair_j[NPAIRS];

    const int tid  = threadIdx.x;       // 0..255
    const int lane = tid & 31;
    const int wave = tid >> 5;          // 0..7 (8 wave32s)
    const int seg  = lane >> 4;         // 0 = lanes 0-15, 1 = lanes 16-31
    const int l15  = lane & 15;
    const int b    = blockIdx.x;        // batch element

    // --- stage W into LDS (4096 floats; regular temporal: reused by all blocks via L2) ---
    {
        const v4f* Wv  = (const v4f*)W;
        v4f*       lWv = (v4f*)lds_W;
        #pragma unroll
        for (int k = tid; k < (NE * NE) / 4; k += 256) lWv[k] = Wv[k];
    }
    // --- stage x[b] into LDS (2048 floats; non-temporal: each x[b] read exactly once) ---
    {
        const v4f* xv  = (const v4f*)(x + (size_t)b * NF * NE);
        v4f*       lxv = (v4f*)lds_x;
        #pragma unroll
        for (int k = tid; k < (NF * NE) / 4; k += 256)
            lxv[k] = __builtin_nontemporal_load(xv + k);
    }
    // --- build (i,j) pair table for p in [0,496) ---
    for (int p = tid; p < NPAIRS; p += 256) {
        int i = 0, base = 0;
        while (base + (NF - 1 - i) <= p) { base += NF - 1 - i; ++i; }
        pair_i[p] = (unsigned short)i;
        pair_j[p] = (unsigned short)(i + 1 + (p - base));
    }
    __syncthreads();

    // --- WMMA: xw[32x64] = x[b][32x64] @ W[64x64] -----------------------------
    // wave -> 16x16 tile: mt in {0,16}, nt in {0,16,32,48}; K-loop over 64 in steps of 4.
    const int mt = (wave >> 2) * 16;
    const int nt = (wave & 3) * 16;

    v8f acc = {};
    #pragma unroll
    for (int k = 0; k < NE; k += 4) {
        // A fragment (16x4): lanes 0-15 hold K=k,k+1; lanes 16-31 hold K=k+2,k+3
        v2f afrag;
        const int arow = mt + l15;
        const int ak   = k + seg * 2;
        afrag.x = lds_x[arow * NE + ak];
        afrag.y = lds_x[arow * NE + ak + 1];
        // B fragment (4x16): lanes 0-15 hold K=k,k+1; lanes 16-31 hold K=k+2,k+3
        v2f bfrag;
        const int bcol = nt + l15;
        const int bk   = k + seg * 2;
        bfrag.x = lds_W[bk * NE + bcol];
        bfrag.y = lds_W[(bk + 1) * NE + bcol];
        // D = A x B + C  -> v_wmma_f32_16x16x4_f32
        acc = __builtin_amdgcn_wmma_f32_16x16x4_f32(
            /*neg_a=*/false, afrag, /*neg_b=*/false, bfrag,
            /*c_mod=*/(short)0, acc, /*reuse_a=*/false, /*reuse_b=*/false);
    }
    // C/D layout: VGPR v, lanes 0-15 -> row mt+v, lanes 16-31 -> row mt+8+v; col nt+l15
    #pragma unroll
    for (int v = 0; v < 8; ++v) {
        lds_xw[(mt + v + seg * 8) * NE + nt + l15] = acc[v];
    }
    __syncthreads();

    // --- pair products: out[b,p,:] = xw[i_p,:] * x[j_p,:] ---------------------
    // 496 pairs = 31 iters * (8 waves * 2 pairs); 16 lanes x float4 cover one 64-float row.
    // Stores are non-temporal: output is write-once, never re-read.
    float* outb = out + (size_t)b * NPAIRS * NE;
    #pragma unroll 4
    for (int it = 0; it < NPAIRS / 16; ++it) {
        const int p = it * 16 + wave * 2 + seg;
        const int i = pair_i[p];
        const int j = pair_j[p];
        const v4f a4 = ((const v4f*)(lds_xw + i * NE))[l15];
        const v4f b4 = ((const v4f*)(lds_x  + j * NE))[l15];
        const v4f r  = a4 * b4;
        __builtin_nontemporal_store(r, (v4f*)(outb + (size_t)p * NE) + l15);
    }
}

extern "C" void kernel_launch(void* const* d_in, const int* in_sizes, int n_in,
                              void* d_out, int out_size, void* d_ws, size_t ws_size,
                              hipStream_t stream)
{
    const float* x = (const float*)d_in[0];   // [B, 32, 64] fp32
    const float* W = (const float*)d_in[1];   // [64, 64]    fp32
    float*     out = (float*)d_out;           // [B, 496, 64] fp32

    const int B = in_sizes[0] / (NF * NE);    // 8192
    bilinear_wmma_kernel<<<B, 256, 0, stream>>>(x, W, out);
}